// QVAE_CF_41755672051861
// MI455X (gfx1250) — compile-verified
//
#include <hip/hip_runtime.h>
#include <cstdint>
#include <cstddef>

// Problem constants (match reference)
#define B_SZ   32768
#define D_DIM  512
#define P_DIM  8
#define C_DIM  512
#define DC_DIM 64
#define TILE_M 32

typedef float v2f __attribute__((ext_vector_type(2)));
typedef float v8f __attribute__((ext_vector_type(8)));

__device__ __forceinline__ uint32_t pcg_hash(uint32_t x) {
    x = x * 747796405u + 2891336453u;
    uint32_t w = ((x >> ((x >> 28u) + 4u)) ^ x) * 277803737u;
    return (w >> 22u) ^ w;
}

__device__ __forceinline__ float u01(uint32_t h) {
    // uniform in (0, 1), safe for log(log())
    return fminf(fmaxf((h >> 8) * (1.0f / 16777216.0f), 1e-9f), 0.999999f);
}

__device__ __forceinline__ float gaussian(uint32_t s) {
    float u1 = u01(pcg_hash(s));
    float u2 = u01(pcg_hash(s ^ 0x85ebca6bu));
    return sqrtf(-2.0f * __logf(u1)) * __cosf(6.28318530718f * u2);
}

// ---------------------------------------------------------------------------
// Kernel 0: c2[p][c] = sum_d centroids[p][c][d]^2
// ---------------------------------------------------------------------------
__global__ void qvae_c2_kernel(const float* __restrict__ centroids,
                               float* __restrict__ c2) {
    int idx = blockIdx.x * blockDim.x + threadIdx.x;   // [0, P*C)
    if (idx >= P_DIM * C_DIM) return;
    const float* row = centroids + (size_t)idx * DC_DIM;
    float s = 0.f;
#pragma unroll
    for (int d = 0; d < DC_DIM; d += 4) {
        float4 v = *reinterpret_cast<const float4*>(row + d);
        s = fmaf(v.x, v.x, s);
        s = fmaf(v.y, v.y, s);
        s = fmaf(v.z, v.z, s);
        s = fmaf(v.w, v.w, s);
    }
    c2[idx] = s;
}

// ---------------------------------------------------------------------------
// Kernel 1: per (32-row tile, partition):
//   cross = u @ centroids^T via V_WMMA_F32_16X16X4_F32
//   Each wave owns 64 columns x 32 rows = 2 row-tiles x 4 col-tiles.
//   B fragments are reused across both row-tiles (halves L2 traffic).
//   neg_dist -> softmax -> logits (non-temporal) ; gumbel argmax -> code
// ---------------------------------------------------------------------------
__global__ __launch_bounds__(256) void qvae_encode_kernel(
    const float* __restrict__ user_emb,
    const float* __restrict__ centroids,
    const int*   __restrict__ user_id,
    const float* __restrict__ c2,
    float* __restrict__ logits,     // [B][P][C]
    int*   __restrict__ code) {     // [B][P]
    const int p    = blockIdx.y;
    const int row0 = blockIdx.x * TILE_M;
    const int tid  = threadIdx.x;
    const int wave = tid >> 5;
    const int lane = tid & 31;

    __shared__ float sU[TILE_M][DC_DIM];        // 8 KB   user tile
    __shared__ float sND[TILE_M][C_DIM + 8];    // 66 KB  neg_dist (padded)
    __shared__ float sP0[TILE_M][8];
    __shared__ float sP1[TILE_M][8];
    __shared__ int   sPi[TILE_M][8];
    __shared__ float sU2[TILE_M];
    __shared__ float sRowMax[TILE_M];
    __shared__ float sRowSum[TILE_M];

    // ---- stage user tile: 32 rows x 64 floats = 512 float4, 2 per thread
#pragma unroll
    for (int pass = 0; pass < 2; ++pass) {
        const int flat = (tid + pass * 256) * 4;     // float index
        const int r    = flat >> 6;                  // row 0..31
        const int c4   = flat & 63;                  // col 0..60 step 4
        const int uid  = user_id[row0 + r];
        const float4 v = *reinterpret_cast<const float4*>(
            user_emb + (size_t)uid * D_DIM + p * DC_DIM + c4);
        *reinterpret_cast<float4*>(&sU[r][c4]) = v;
    }
    __syncthreads();

    if (tid < TILE_M) {
        float s = 0.f;
#pragma unroll
        for (int d = 0; d < DC_DIM; ++d) s = fmaf(sU[tid][d], sU[tid][d], s);
        sU2[tid] = s;
    }
    __syncthreads();

    // ---- WMMA: acc[rt][t] = 16x16 tile (rows rt*16.., cols wave*64+t*16..)
    v8f acc[2][4];
#pragma unroll
    for (int rt = 0; rt < 2; ++rt)
#pragma unroll
        for (int t = 0; t < 4; ++t)
#pragma unroll
            for (int i = 0; i < 8; ++i) acc[rt][t][i] = 0.f;

    const int m  = lane & 15;   // A: row m ; B/C/D: col n
    const int kh = lane >> 4;   // K-half selector
    const float* cpart = centroids + (size_t)p * C_DIM * DC_DIM;

    for (int k0 = 0; k0 < DC_DIM; k0 += 4) {
        const int ka = k0 + 2 * kh;
        // A fragments 16x4 f32 for both row-tiles
        v2f a0, a1;
        a0.x = sU[m][ka];
        a0.y = sU[m][ka + 1];
        a1.x = sU[m + 16][ka];
        a1.y = sU[m + 16][ka + 1];
#pragma unroll
        for (int t = 0; t < 4; ++t) {
            const int c = (wave << 6) + (t << 4) + m;     // centroid index
            const float2 bv = *reinterpret_cast<const float2*>(
                cpart + (size_t)c * DC_DIM + ka);
            v2f bfrag;
            bfrag.x = bv.x;
            bfrag.y = bv.y;
            acc[0][t] = __builtin_amdgcn_wmma_f32_16x16x4_f32(
                false, a0, false, bfrag, (short)0, acc[0][t], false, false);
            acc[1][t] = __builtin_amdgcn_wmma_f32_16x16x4_f32(
                false, a1, false, bfrag, (short)0, acc[1][t], false, false);
        }
    }

    // ---- neg_dist = -sqrt(max(u2 + c2 - 2*cross, 0) + eps) into LDS
#pragma unroll
    for (int t = 0; t < 4; ++t) {
        const int c = (wave << 6) + (t << 4) + m;
        const float cc2 = c2[p * C_DIM + c];
#pragma unroll
        for (int rt = 0; rt < 2; ++rt)
#pragma unroll
            for (int r = 0; r < 8; ++r) {
                const int mm = (rt << 4) + r + (kh << 3);  // C/D: M = reg+8*half
                float d2 = sU2[mm] + cc2 - 2.0f * acc[rt][t][r];
                d2 = fmaxf(d2, 0.0f);
                sND[mm][c] = -sqrtf(d2 + 1e-12f);
            }
    }
    __syncthreads();

    // ---- softmax over C + gumbel argmax; 8 threads per row, cols j + 8k
    const int r = tid >> 3;     // row 0..31
    const int j = tid & 7;
    const int b = row0 + r;

    float lmax = -3.4e38f, gmax = -3.4e38f;
    int gidx = 0;
    for (int k = 0; k < 64; ++k) {
        const int c = j + (k << 3);
        const float nd = sND[r][c];
        lmax = fmaxf(lmax, nd);
        const uint32_t h = pcg_hash((uint32_t)b * (uint32_t)(P_DIM * C_DIM) +
                                    (uint32_t)(p * C_DIM + c) + 0x9e3779b9u);
        const float g = -__logf(-__logf(u01(h)));   // gumbel noise
        const float y = nd + g;                     // TAU = 1
        if (y > gmax) { gmax = y; gidx = c; }
    }
    sP0[r][j] = lmax;
    sP1[r][j] = gmax;
    sPi[r][j] = gidx;
    __syncthreads();
    if (j == 0) {
        float rm = sP0[r][0], gm = sP1[r][0];
        int gi = sPi[r][0];
#pragma unroll
        for (int q = 1; q < 8; ++q) {
            rm = fmaxf(rm, sP0[r][q]);
            if (sP1[r][q] > gm) { gm = sP1[r][q]; gi = sPi[r][q]; }
        }
        sRowMax[r] = rm;
        code[b * P_DIM + p] = gi;
    }
    __syncthreads();

    const float rmax = sRowMax[r];
    float lsum = 0.f;
    for (int k = 0; k < 64; ++k) {
        const int c = j + (k << 3);
        lsum += __expf(sND[r][c] - rmax);
    }
    sP0[r][j] = lsum;
    __syncthreads();
    if (j == 0) {
        float s = 0.f;
#pragma unroll
        for (int q = 0; q < 8; ++q) s += sP0[r][q];
        sRowSum[r] = s;
    }
    __syncthreads();

    // streaming (non-temporal) logits write: never re-read, keep L2 for
    // centroid reuse instead of 537 MB of dead write data
    const float rinv = 1.0f / sRowSum[r];
    float* lrow = logits + ((size_t)b * P_DIM + p) * C_DIM;
    for (int k = 0; k < 64; ++k) {
        const int c = j + (k << 3);
        __builtin_nontemporal_store(__expf(sND[r][c] - rmax) * rinv, &lrow[c]);
    }
}

// ---------------------------------------------------------------------------
// Kernel 2: one wave per b:
//   score[b] = sum_p dot(centroids[p][code[b][p]][:], mu + eps*exp(0.5*logvar))
// ---------------------------------------------------------------------------
__global__ __launch_bounds__(256) void qvae_score_kernel(
    const float* __restrict__ centroids,
    const float* __restrict__ item_mu,
    const float* __restrict__ item_logvar,
    const int*   __restrict__ item_id,
    const int*   __restrict__ code,
    float* __restrict__ score) {
    const int b    = blockIdx.x * 8 + (threadIdx.x >> 5);
    const int lane = threadIdx.x & 31;
    const int iid  = item_id[b];
    const float* mu = item_mu     + (size_t)iid * D_DIM;
    const float* lv = item_logvar + (size_t)iid * D_DIM;

    float accv = 0.f;
#pragma unroll
    for (int p = 0; p < P_DIM; ++p) {
        const int c = code[b * P_DIM + p];
        const float* cen = centroids + ((size_t)p * C_DIM + c) * DC_DIM;
#pragma unroll
        for (int h = 0; h < 2; ++h) {
            const int d  = lane + (h << 5);     // 0..63 within partition
            const int gd = p * DC_DIM + d;      // 0..511 global dim
            const float eps = gaussian((uint32_t)b * (uint32_t)D_DIM +
                                       (uint32_t)gd + 0x2545f491u);
            // mu/logvar are read exactly once -> non-temporal loads
            const float muv = __builtin_nontemporal_load(&mu[gd]);
            const float lvv = __builtin_nontemporal_load(&lv[gd]);
            const float iv  = fmaf(eps, __expf(0.5f * lvv), muv);
            accv = fmaf(cen[d], iv, accv);
        }
    }
#pragma unroll
    for (int off = 16; off > 0; off >>= 1)
        accv += __shfl_xor(accv, off, 32);
    if (lane == 0) score[b] = accv;
}

// ---------------------------------------------------------------------------
extern "C" void kernel_launch(void* const* d_in, const int* in_sizes, int n_in,
                              void* d_out, int out_size, void* d_ws, size_t ws_size,
                              hipStream_t stream) {
    const float* user_emb    = (const float*)d_in[0];
    const float* centroids   = (const float*)d_in[1];
    const float* item_mu     = (const float*)d_in[2];
    const float* item_logvar = (const float*)d_in[3];
    const int*   user_id     = (const int*)d_in[4];
    const int*   item_id     = (const int*)d_in[5];

    float* score  = (float*)d_out;          // [B]
    float* logits = (float*)d_out + B_SZ;   // [B][P][C]

    float* c2    = (float*)d_ws;            // 16 KB
    int*   codew = (int*)((char*)d_ws +
                   ((P_DIM * C_DIM * sizeof(float) + 255) & ~(size_t)255));

    qvae_c2_kernel<<<(P_DIM * C_DIM + 255) / 256, 256, 0, stream>>>(centroids, c2);

    dim3 grid1(B_SZ / TILE_M, P_DIM);
    qvae_encode_kernel<<<grid1, 256, 0, stream>>>(user_emb, centroids, user_id,
                                                  c2, logits, codew);

    qvae_score_kernel<<<B_SZ / 8, 256, 0, stream>>>(centroids, item_mu,
                                                    item_logvar, item_id,
                                                    codew, score);
}